// MLP_66778151518217
// MI455X (gfx1250) — compile-verified
//
#include <hip/hip_runtime.h>
#include <hip/hip_bf16.h>

typedef float v2f __attribute__((ext_vector_type(2)));
typedef float v8f __attribute__((ext_vector_type(8)));

#define NN 100000
#define DD 64
#define EE 1200000

// ---------------------------------------------------------------------------
// Kernel 1: fused Q/K/V/skip GEMM using V_WMMA_F32_16X16X4_F32.
// Block = 128 threads = 4 waves; wave w computes x_tile @ W_w + b_w for one
// of {Wq,Wk,Wv,Ws}. 16 node rows per block (100000 % 16 == 0 -> no ragged
// tiles, EXEC stays all-ones around WMMA as required).
// ---------------------------------------------------------------------------
__global__ __launch_bounds__(128) void qkvs_wmma_kernel(
    const float* __restrict__ x,
    const float* __restrict__ Wq, const float* __restrict__ bq,
    const float* __restrict__ Wk, const float* __restrict__ bk,
    const float* __restrict__ Wv, const float* __restrict__ bv,
    const float* __restrict__ Ws, const float* __restrict__ bs,
    float* __restrict__ qo, float* __restrict__ ko,
    float* __restrict__ vo, float* __restrict__ so)
{
    // 16 rows x 64 cols, padded to 68 floats/row so the column reads
    // (stride 68 -> bank 4*row + col) hit distinct LDS banks.
    __shared__ float xs[16 * 68];

    const int tid = threadIdx.x;
    const int m0  = blockIdx.x * 16;

    for (int i = tid; i < 16 * 64; i += 128) {
        const int r = i >> 6, c = i & 63;
        xs[r * 68 + c] = x[(size_t)(m0 + r) * DD + c];
    }
    __syncthreads();

    const int wid  = tid >> 5;
    const int lane = tid & 31;
    const int half = lane >> 4;   // 0: lanes 0-15, 1: lanes 16-31
    const int l15  = lane & 15;

    const float* W;  const float* bias;  float* outp;
    if      (wid == 0) { W = Wq; bias = bq; outp = qo; }
    else if (wid == 1) { W = Wk; bias = bk; outp = ko; }
    else if (wid == 2) { W = Wv; bias = bv; outp = vo; }
    else               { W = Ws; bias = bs; outp = so; }

    for (int n0 = 0; n0 < DD; n0 += 16) {
        // C/D layout: element i of acc is (row = i + 8*half, col = n0 + l15);
        // bias is per-column so every acc element starts at bias[col].
        const float bval = bias[n0 + l15];
        v8f acc;
        #pragma unroll
        for (int i = 0; i < 8; ++i) acc[i] = bval;

        #pragma unroll
        for (int kk = 0; kk < DD; kk += 4) {
            // A (16x4 f32): lanes 0-15 hold K=kk,kk+1 of row M=l15;
            //               lanes 16-31 hold K=kk+2,kk+3 of row M=l15.
            v2f a, b;
            a.x = xs[l15 * 68 + kk + 2 * half + 0];
            a.y = xs[l15 * 68 + kk + 2 * half + 1];
            // B (4x16 f32): mirrored layout, W row-major [k][n].
            b.x = W[(kk + 2 * half + 0) * DD + n0 + l15];
            b.y = W[(kk + 2 * half + 1) * DD + n0 + l15];
            acc = __builtin_amdgcn_wmma_f32_16x16x4_f32(
                /*neg_a=*/false, a, /*neg_b=*/false, b,
                /*c_mod=*/(short)0, acc, /*reuse_a=*/false, /*reuse_b=*/false);
        }

        #pragma unroll
        for (int i = 0; i < 8; ++i) {
            const int row = i + 8 * half;
            outp[(size_t)(m0 + row) * DD + n0 + l15] = acc[i];
        }
    }
}

// ---------------------------------------------------------------------------
// Kernel 2: initialize reduction buffers (node max = -inf, denom = 0, agg = 0)
// ---------------------------------------------------------------------------
__global__ __launch_bounds__(256) void init_kernel(float* __restrict__ nmax,
                                                   float* __restrict__ denom,
                                                   float* __restrict__ agg)
{
    const int i = blockIdx.x * 256 + threadIdx.x;
    if (i < NN) { nmax[i] = -INFINITY; denom[i] = 0.0f; }
    if (i < NN * DD) agg[i] = 0.0f;
}

// ---------------------------------------------------------------------------
// Kernel 3: per-edge attention logits: (q[dst] . k[src]) / sqrt(D)
// One thread per edge: two contiguous 256B row reads (L2-resident).
// ---------------------------------------------------------------------------
__global__ __launch_bounds__(256) void score_kernel(
    const float* __restrict__ q, const float* __restrict__ k,
    const int* __restrict__ ei, float* __restrict__ scores)
{
    const int e = blockIdx.x * 256 + threadIdx.x;
    if (e >= EE) return;
    const int s = ei[e];          // source j
    const int d = ei[EE + e];     // target i
    const float4* qp = (const float4*)(q + (size_t)d * DD);
    const float4* kp = (const float4*)(k + (size_t)s * DD);
    float acc = 0.0f;
    #pragma unroll
    for (int i = 0; i < 16; ++i) {
        const float4 a = qp[i], b = kp[i];
        acc += a.x * b.x + a.y * b.y + a.z * b.z + a.w * b.w;
    }
    scores[e] = acc * 0.125f;     // 1/sqrt(64)
}

// ---------------------------------------------------------------------------
// Kernel 4: segment max via float-as-int atomic max
// ---------------------------------------------------------------------------
__device__ __forceinline__ void atomicMaxF(float* addr, float v)
{
    if (v >= 0.0f) atomicMax((int*)addr, __float_as_int(v));
    else           atomicMin((unsigned int*)addr, __float_as_uint(v));
}

__global__ __launch_bounds__(256) void segmax_kernel(
    const float* __restrict__ scores, const int* __restrict__ ei,
    float* __restrict__ nmax)
{
    const int e = blockIdx.x * 256 + threadIdx.x;
    if (e >= EE) return;
    atomicMaxF(&nmax[ei[EE + e]], scores[e]);
}

// ---------------------------------------------------------------------------
// Kernel 5: ex = exp(score - max[dst]); denom[dst] += ex
// ---------------------------------------------------------------------------
__global__ __launch_bounds__(256) void expsum_kernel(
    const float* __restrict__ scores, const int* __restrict__ ei,
    const float* __restrict__ nmax, float* __restrict__ ex,
    float* __restrict__ denom)
{
    const int e = blockIdx.x * 256 + threadIdx.x;
    if (e >= EE) return;
    const int d = ei[EE + e];
    const float v = __expf(scores[e] - nmax[d]);
    ex[e] = v;
    atomicAdd(&denom[d], v);
}

// ---------------------------------------------------------------------------
// Kernel 6: agg[dst] += (ex/denom[dst]) * v[src]  -- 64 threads per edge
// ---------------------------------------------------------------------------
__global__ __launch_bounds__(256) void agg_kernel(
    const float* __restrict__ v, const int* __restrict__ ei,
    const float* __restrict__ ex, const float* __restrict__ denom,
    float* __restrict__ agg)
{
    const int idx = blockIdx.x * 256 + threadIdx.x;   // < E*64 = 76.8M, fits int
    const int e = idx >> 6;
    const int d = idx & 63;
    if (e >= EE) return;
    const int s = ei[e];
    const int t = ei[EE + e];
    const float alpha = ex[e] / denom[t];
    atomicAdd(&agg[(size_t)t * DD + d], alpha * v[(size_t)s * DD + d]);
}

// ---------------------------------------------------------------------------
// Kernel 7: out = relu(agg + skip)
// ---------------------------------------------------------------------------
__global__ __launch_bounds__(256) void finish_kernel(
    float* __restrict__ out, const float* __restrict__ skip)
{
    const int i = blockIdx.x * 256 + threadIdx.x;
    if (i >= NN * DD) return;
    out[i] = fmaxf(out[i] + skip[i], 0.0f);
}

// ---------------------------------------------------------------------------
extern "C" void kernel_launch(void* const* d_in, const int* in_sizes, int n_in,
                              void* d_out, int out_size, void* d_ws, size_t ws_size,
                              hipStream_t stream)
{
    const float* x   = (const float*)d_in[0];
    const int*   ei  = (const int*)  d_in[1];
    const float* Wq  = (const float*)d_in[2];
    const float* bq  = (const float*)d_in[3];
    const float* Wk  = (const float*)d_in[4];
    const float* bk  = (const float*)d_in[5];
    const float* Wv  = (const float*)d_in[6];
    const float* bv  = (const float*)d_in[7];
    const float* Ws  = (const float*)d_in[8];
    const float* bs  = (const float*)d_in[9];
    float* out = (float*)d_out;

    // Workspace layout (floats)
    float* ws     = (float*)d_ws;
    float* qbuf   = ws;                               // N*D
    float* kbuf   = qbuf  + (size_t)NN * DD;          // N*D
    float* vbuf   = kbuf  + (size_t)NN * DD;          // N*D
    float* sbuf   = vbuf  + (size_t)NN * DD;          // N*D (skip = x@Ws + bs)
    float* scores = sbuf  + (size_t)NN * DD;          // E
    float* ex     = scores + (size_t)EE;              // E
    float* nmax   = ex     + (size_t)EE;              // N
    float* denom  = nmax   + (size_t)NN;              // N

    // 1) Q/K/V/skip projections via fp32 WMMA (6250 tiles of 16 rows)
    qkvs_wmma_kernel<<<NN / 16, 128, 0, stream>>>(
        x, Wq, bq, Wk, bk, Wv, bv, Ws, bs, qbuf, kbuf, vbuf, sbuf);

    // 2) init reductions + zero the aggregation accumulator (d_out)
    init_kernel<<<(NN * DD + 255) / 256, 256, 0, stream>>>(nmax, denom, out);

    // 3) per-edge scores
    score_kernel<<<(EE + 255) / 256, 256, 0, stream>>>(qbuf, kbuf, ei, scores);

    // 4) segment max
    segmax_kernel<<<(EE + 255) / 256, 256, 0, stream>>>(scores, ei, nmax);

    // 5) exp + segment sum
    expsum_kernel<<<(EE + 255) / 256, 256, 0, stream>>>(scores, ei, nmax, ex, denom);

    // 6) weighted aggregation into d_out
    agg_kernel<<<(EE * DD) / 256, 256, 0, stream>>>(vbuf, ei, ex, denom, out);

    // 7) skip + ReLU
    finish_kernel<<<(NN * DD + 255) / 256, 256, 0, stream>>>(out, sbuf);
}